// DualThresholdSelfregulatingIntegrate_88854283420100
// MI455X (gfx1250) — compile-verified
//
#include <hip/hip_runtime.h>
#include <cstdint>
#include <math.h>

// Problem shape (fixed by the reference): inputs [B, T, D] fp32, output = spikes.
#define BDIM   64
#define TDIM   256
#define DDIM   1024
#define SLICE  256                 // d-channels per block (== blockDim.x, 8 wave32)
#define NSLICE (DDIM / SLICE)      // 4 blocks per batch row -> 256 blocks total
#define DEPTH  32                  // async pipeline stages resident in LDS
#define SLOT_BYTES (SLICE * 4)     // 1 KB per stage per block

// gfx1250 async copy, GVS mode: addr = SGPR64_base + VGPR32_off (+IOFFSET).
// vdst = LDS byte address (VGPR), tracked by ASYNCcnt.
#define ASYNC_LOAD_GVS(ldsaddr, voff, sbase)                             \
  asm volatile("global_load_async_to_lds_b32 %0, %1, %2"                 \
               :: "v"(ldsaddr), "v"(voff), "s"(sbase) : "memory")

// s_wait_asynccnt <= n  (memory clobber pins LDS reads after the wait)
#define WAIT_ASYNC(nlit) asm volatile("s_wait_asynccnt " #nlit ::: "memory")

// gfx1250 hardware tanh (TRANS unit). ISA: one independent op / V_NOP must
// separate a trans op from the first use of its result -> bundle a v_nop.
__device__ __forceinline__ float hw_tanh(float x) {
  float r;
  asm("v_tanh_f32 %0, %1\n\tv_nop" : "=v"(r) : "v"(x));
  return r;
}

__global__ __launch_bounds__(SLICE)
void snn_scan_async_kernel(const float* __restrict__ inp, float* __restrict__ out) {
  __shared__ float stage[DEPTH * SLICE];   // 32 KB of the 320 KB/WGP pool

  const int tid = threadIdx.x;
  const int b   = blockIdx.x >> 2;         // NSLICE == 4
  const int sl  = blockIdx.x & (NSLICE - 1);
  const int d   = sl * SLICE + tid;

  // Uniform (block-scalar) batch-row base -> SGPR pair for GVS addressing.
  const float* sbase = inp + (size_t)b * TDIM * DDIM;
  float*       o     = out + (size_t)b * TDIM * DDIM + d;

  // Flat pointer to LDS: low 32 bits == LDS byte offset (aperture layout),
  // which is exactly what the async instruction's VDST address VGPR wants.
  const unsigned lds0 = (unsigned)(uintptr_t)(&stage[tid]);

  // Per-lane 32-bit global offset: (t*DDIM + d)*4, advances by 4096 B per step.
  unsigned voff = (unsigned)(d * 4);

  // ---- prologue: fill the DEPTH-stage pipeline ----
  #pragma unroll
  for (int k = 0; k < DEPTH; ++k) {
    ASYNC_LOAD_GVS(lds0 + (unsigned)(k * SLOT_BYTES), voff, sbase);
    voff += (unsigned)(DDIM * 4);
  }

  float v = 0.0f;                              // membrane state

  // ---- steady state: wait oldest stage, consume, refill same slot ----
  #pragma unroll 4
  for (int t = 0; t < TDIM - DEPTH; ++t) {
    WAIT_ASYNC(31);                            // DEPTH-1 -> oldest stage landed
    const int slot = t & (DEPTH - 1);
    float x = stage[slot * SLICE + tid];       // ds_load_b32 (own lane's word)

    float r = hw_tanh(x);                      // v_tanh_f32
    v       = v + r * 0.01f;                   // integrate (DT = 0.01)
    float s = (v >=  1.0f) ? 1.0f : 0.0f;      // signed spike in {-1, 0, +1}
    s       = (v <= -1.0f) ? -1.0f : s;
    v      -= s;                               // subtractive reset
    __builtin_nontemporal_store(s * 100.0f, o);  // (sp-sn)/DT, keep out of L2

    ASYNC_LOAD_GVS(lds0 + (unsigned)(slot * SLOT_BYTES), voff, sbase);
    voff += (unsigned)(DDIM * 4);
    o    += DDIM;
  }

  // ---- drain: all stages issued; wait for everything, consume the tail ----
  WAIT_ASYNC(0);
  #pragma unroll 4
  for (int t = TDIM - DEPTH; t < TDIM; ++t) {
    const int slot = t & (DEPTH - 1);
    float x = stage[slot * SLICE + tid];
    float r = hw_tanh(x);
    v       = v + r * 0.01f;
    float s = (v >=  1.0f) ? 1.0f : 0.0f;
    s       = (v <= -1.0f) ? -1.0f : s;
    v      -= s;
    __builtin_nontemporal_store(s * 100.0f, o);
    o += DDIM;
  }
}

extern "C" void kernel_launch(void* const* d_in, const int* in_sizes, int n_in,
                              void* d_out, int out_size, void* d_ws, size_t ws_size,
                              hipStream_t stream) {
  const float* inp = (const float*)d_in[0];
  float*       out = (float*)d_out;
  (void)in_sizes; (void)n_in; (void)out_size; (void)d_ws; (void)ws_size;

  dim3 grid(BDIM * NSLICE);   // 256 blocks
  dim3 block(SLICE);          // 256 threads = 8 wave32
  snn_scan_async_kernel<<<grid, block, 0, stream>>>(inp, out);
}